// MMDLoss_8675833938587
// MI455X (gfx1250) — compile-verified
//
#include <hip/hip_runtime.h>
#include <hip/hip_bf16.h>

typedef __attribute__((ext_vector_type(2))) float v2f;
typedef __attribute__((ext_vector_type(8))) float v8f;

#define N_TOT   8192
#define BS      4096
#define D_DIM   256
#define WBLK    64      // 64x64 output per wave
#define TPB     256     // 8 waves per block

// ---------------------------------------------------------------------------
// Kernel 1: row squared-norms of virtual concat(source,target); also zero the
// 5 double accumulators in workspace (sum_d2, xx, xy, yx, yy).
// ---------------------------------------------------------------------------
__global__ __launch_bounds__(TPB) void mmd_sqnorm_kernel(
    const float* __restrict__ src, const float* __restrict__ tgt,
    float* __restrict__ sq, double* __restrict__ accs) {
  if (blockIdx.x == 0 && threadIdx.x < 5) accs[threadIdx.x] = 0.0;

  const int lane = threadIdx.x & 31;
  const int row  = (blockIdx.x * TPB + threadIdx.x) >> 5;  // one row per wave
  const float* p = (row < BS) ? (src + (size_t)row * D_DIM)
                              : (tgt + (size_t)(row - BS) * D_DIM);
  // lane covers 8 consecutive floats (two float4 loads)
  const float4* p4 = (const float4*)(p + lane * 8);
  float4 x0 = p4[0];
  float4 x1 = p4[1];
  float s = x0.x * x0.x + x0.y * x0.y + x0.z * x0.z + x0.w * x0.w +
            x1.x * x1.x + x1.y * x1.y + x1.z * x1.z + x1.w * x1.w;
#pragma unroll
  for (int off = 16; off > 0; off >>= 1) s += __shfl_down(s, off);
  if (lane == 0) sq[row] = s;
}

// ---------------------------------------------------------------------------
// Kernel 2/3: tiled GEMM  gram = T*T^T  via V_WMMA_F32_16X16X4_F32.
// MODE 0: accumulate sum(max(d2,0)) into accs[0] (bandwidth numerator).
// MODE 1: accumulate sum over 5 RBF scales into the quadrant accumulator.
// Each wave computes a 64x64 block: acc[4][4] 16x16 f32 tiles.
// ---------------------------------------------------------------------------
template <int MODE>
__global__ __launch_bounds__(TPB, 1) void mmd_gemm_kernel(
    const float* __restrict__ src, const float* __restrict__ tgt,
    const float* __restrict__ sq, double* __restrict__ accs) {
  const int lane   = threadIdx.x & 31;
  const int waveId = (blockIdx.x << 3) | (threadIdx.x >> 5);  // 0..16383
  const int bi = waveId >> 7;   // 0..127 row block
  const int bj = waveId & 127;  // 0..127 col block
  const int row0 = bi << 6;
  const int col0 = bj << 6;

  const float* Ab = (row0 < BS) ? (src + (size_t)row0 * D_DIM)
                                : (tgt + (size_t)(row0 - BS) * D_DIM);
  const float* Bb = (col0 < BS) ? (src + (size_t)col0 * D_DIM)
                                : (tgt + (size_t)(col0 - BS) * D_DIM);

  const int ml = lane & 15;   // row (A) / col (B) within 16-tile
  const int hi = lane >> 4;   // selects K pair {0,1} vs {2,3}

  // Per-tile row pointers: lane loads float2 = T[tile*16+ml][k0 + 2*hi .. +1]
  const float* Arow[4];
  const float* Brow[4];
#pragma unroll
  for (int t = 0; t < 4; ++t) {
    Arow[t] = Ab + (size_t)(t * 16 + ml) * D_DIM + 2 * hi;
    Brow[t] = Bb + (size_t)(t * 16 + ml) * D_DIM + 2 * hi;
  }

  v8f acc[4][4] = {};

#pragma unroll 2
  for (int k0 = 0; k0 < D_DIM; k0 += 4) {
    v2f a[4], b[4];
#pragma unroll
    for (int t = 0; t < 4; ++t) {
      a[t] = *(const v2f*)(Arow[t] + k0);
      b[t] = *(const v2f*)(Brow[t] + k0);
    }
#pragma unroll
    for (int tm = 0; tm < 4; ++tm) {
#pragma unroll
      for (int tn = 0; tn < 4; ++tn) {
        acc[tm][tn] = __builtin_amdgcn_wmma_f32_16x16x4_f32(
            /*neg_a=*/false, a[tm], /*neg_b=*/false, b[tn],
            /*c_mod=*/(short)0, acc[tm][tn],
            /*reuse_a=*/false, /*reuse_b=*/false);
      }
    }
  }

  // --- Epilogue ---------------------------------------------------------
  // C tile layout: VGPR r, lanes 0-15 -> M=r, N=lane; lanes 16-31 -> M=8+r.
  float s16 = 0.0f;
  if (MODE == 1) {
    const double S  = accs[0];
    const double Nd = (double)N_TOT;
    const double bw = S / (Nd * Nd - Nd) / 4.0;  // / KERNEL_MUL^(NUM/2)
    s16 = (float)(1.0 / (16.0 * bw));            // exponent scale for e
  }

  float sqa[4][8];
#pragma unroll
  for (int tm = 0; tm < 4; ++tm) {
    const int base_i = row0 + tm * 16 + (hi << 3);
#pragma unroll
    for (int r = 0; r < 8; ++r) sqa[tm][r] = sq[base_i + r];
  }
  float sqb[4];
#pragma unroll
  for (int tn = 0; tn < 4; ++tn) sqb[tn] = sq[col0 + tn * 16 + ml];

  float lsum = 0.0f;
#pragma unroll
  for (int tm = 0; tm < 4; ++tm) {
#pragma unroll
    for (int tn = 0; tn < 4; ++tn) {
      const v8f g = acc[tm][tn];
#pragma unroll
      for (int r = 0; r < 8; ++r) {
        float d2 = sqa[tm][r] + sqb[tn] - 2.0f * g[r];
        d2 = fmaxf(d2, 0.0f);
        if (MODE == 0) {
          lsum += d2;
        } else {
          // sum_{i=0..4} exp(-d2/(bw*2^i)) = e + e^2 + e^4 + e^8 + e^16,
          // e = exp(-d2/(16*bw))
          const float e1  = __expf(-d2 * s16);
          const float e2  = e1 * e1;
          const float e4  = e2 * e2;
          const float e8  = e4 * e4;
          const float e16 = e8 * e8;
          lsum += e1 + e2 + e4 + e8 + e16;
        }
      }
    }
  }

#pragma unroll
  for (int off = 16; off > 0; off >>= 1) lsum += __shfl_down(lsum, off);

  if (lane == 0) {
    if (MODE == 0) {
      atomicAdd(&accs[0], (double)lsum);
    } else {
      const int q = ((row0 >= BS) << 1) | (col0 >= BS);  // 0=xx,1=xy,2=yx,3=yy
      atomicAdd(&accs[1 + q], (double)lsum);
    }
  }
}

// ---------------------------------------------------------------------------
// Kernel 4: finalize  xx + yy - xy - yx  (each quadrant has BS^2 elements)
// ---------------------------------------------------------------------------
__global__ void mmd_finalize_kernel(const double* __restrict__ accs,
                                    float* __restrict__ out) {
  const double inv = 1.0 / ((double)BS * (double)BS);
  out[0] = (float)((accs[1] + accs[4] - accs[2] - accs[3]) * inv);
}

// ---------------------------------------------------------------------------
extern "C" void kernel_launch(void* const* d_in, const int* in_sizes, int n_in,
                              void* d_out, int out_size, void* d_ws,
                              size_t ws_size, hipStream_t stream) {
  const float* src = (const float*)d_in[0];
  const float* tgt = (const float*)d_in[1];
  float* out = (float*)d_out;

  double* accs = (double*)d_ws;                       // 5 doubles
  float* sq    = (float*)((char*)d_ws + 64);          // 8192 floats

  // 8192 rows, one wave each, 8 waves/block
  mmd_sqnorm_kernel<<<N_TOT / 8, TPB, 0, stream>>>(src, tgt, sq, accs);

  // 128x128 wave blocks = 16384 waves, 8 waves/block -> 2048 blocks
  mmd_gemm_kernel<0><<<2048, TPB, 0, stream>>>(src, tgt, sq, accs);
  mmd_gemm_kernel<1><<<2048, TPB, 0, stream>>>(src, tgt, sq, accs);

  mmd_finalize_kernel<<<1, 1, 0, stream>>>(accs, out);
}